// GeometricFeatureExtractor_5540507811898
// MI455X (gfx1250) — compile-verified
//
#include <hip/hip_runtime.h>
#include <hip/hip_bf16.h>
#include <math.h>

typedef __attribute__((ext_vector_type(2))) float v2f;
typedef __attribute__((ext_vector_type(8))) float v8f;

#define NPTS 4096
#define BROWS 4096
#define TPB 256
#define HID 64
#define OUTD 32
#define GEOPAD 16   // geo features padded 10 -> 16 (K multiple of 4/16)
#define HSTRIDE 65  // LDS row stride for hidden tile (bank-conflict pad)

// ---------------- wave32 reductions ----------------
__device__ inline float wred_sum(float v) {
  #pragma unroll
  for (int off = 16; off > 0; off >>= 1) v += __shfl_xor(v, off, 32);
  return v;
}
__device__ inline float wred_min(float v) {
  #pragma unroll
  for (int off = 16; off > 0; off >>= 1) v = fminf(v, __shfl_xor(v, off, 32));
  return v;
}
__device__ inline float wred_max(float v) {
  #pragma unroll
  for (int off = 16; off > 0; off >>= 1) v = fmaxf(v, __shfl_xor(v, off, 32));
  return v;
}

// ---------------------------------------------------------------------------
// Kernel 1: fused geometric feature extraction.
// One block per batch row. Points row (32 KB) staged into LDS during pass-1
// stats; pass 2 (distances from centroid) reads LDS only -> global memory is
// touched exactly once (128 MB stream = the roofline floor).
// ---------------------------------------------------------------------------
__global__ void __launch_bounds__(TPB)
geo_kernel(const float* __restrict__ points, float* __restrict__ geo) {
  __shared__ float2 pts[NPTS];          // 32 KB staging of this row
  __shared__ float pstat[8][8];         // per-wave pass-1 partials
  __shared__ float pdist[8][2];         // per-wave pass-2 partials
  __shared__ float bc[2];               // broadcast centroid

  const int b    = blockIdx.x;
  const int tid  = threadIdx.x;
  const int lane = tid & 31;
  const int wv   = tid >> 5;

  const float4* row4 = (const float4*)(points + (size_t)b * (size_t)NPTS * 2);
  float4* pts4 = (float4*)pts;

  float mnx =  3.4e38f, mny =  3.4e38f;
  float mxx = -3.4e38f, mxy = -3.4e38f;
  float sx = 0.f, sy = 0.f, sxx = 0.f, syy = 0.f;

  // Pass 1: stream global -> LDS, accumulate stats on the fly.
  #pragma unroll
  for (int i = 0; i < (NPTS / 2) / TPB; ++i) {   // 8 iters, float4 = 2 points
    int idx = i * TPB + tid;
    float4 v = row4[idx];
    pts4[idx] = v;
    mnx = fminf(mnx, fminf(v.x, v.z));  mxx = fmaxf(mxx, fmaxf(v.x, v.z));
    mny = fminf(mny, fminf(v.y, v.w));  mxy = fmaxf(mxy, fmaxf(v.y, v.w));
    sx += v.x + v.z;  sy += v.y + v.w;
    sxx += v.x * v.x + v.z * v.z;
    syy += v.y * v.y + v.w * v.w;
  }

  mnx = wred_min(mnx); mxx = wred_max(mxx);
  mny = wred_min(mny); mxy = wred_max(mxy);
  sx  = wred_sum(sx);  sy  = wred_sum(sy);
  sxx = wred_sum(sxx); syy = wred_sum(syy);
  if (lane == 0) {
    pstat[wv][0] = mnx; pstat[wv][1] = mxx;
    pstat[wv][2] = mny; pstat[wv][3] = mxy;
    pstat[wv][4] = sx;  pstat[wv][5] = sy;
    pstat[wv][6] = sxx; pstat[wv][7] = syy;
  }
  __syncthreads();
  if (tid == 0) {
    float a0 = pstat[0][0], a1 = pstat[0][1], a2 = pstat[0][2], a3 = pstat[0][3];
    float a4 = pstat[0][4], a5 = pstat[0][5], a6 = pstat[0][6], a7 = pstat[0][7];
    #pragma unroll
    for (int w = 1; w < 8; ++w) {
      a0 = fminf(a0, pstat[w][0]); a1 = fmaxf(a1, pstat[w][1]);
      a2 = fminf(a2, pstat[w][2]); a3 = fmaxf(a3, pstat[w][3]);
      a4 += pstat[w][4]; a5 += pstat[w][5]; a6 += pstat[w][6]; a7 += pstat[w][7];
    }
    pstat[0][0] = a0; pstat[0][1] = a1; pstat[0][2] = a2; pstat[0][3] = a3;
    pstat[0][4] = a4; pstat[0][5] = a5; pstat[0][6] = a6; pstat[0][7] = a7;
    bc[0] = a4 * (1.0f / NPTS);
    bc[1] = a5 * (1.0f / NPTS);
  }
  __syncthreads();

  const float cx = bc[0], cy = bc[1];

  // Pass 2: distances from centroid, LDS only.
  float s1 = 0.f, s2 = 0.f;
  #pragma unroll
  for (int i = 0; i < NPTS / TPB; ++i) {         // 16 iters
    float2 p = pts[i * TPB + tid];
    float dx = p.x - cx, dy = p.y - cy;
    float d = sqrtf(dx * dx + dy * dy);
    s1 += d; s2 += d * d;
  }
  s1 = wred_sum(s1); s2 = wred_sum(s2);
  if (lane == 0) { pdist[wv][0] = s1; pdist[wv][1] = s2; }
  __syncthreads();

  if (tid == 0) {
    float d1 = 0.f, d2 = 0.f;
    #pragma unroll
    for (int w = 0; w < 8; ++w) { d1 += pdist[w][0]; d2 += pdist[w][1]; }
    const float invN = 1.0f / NPTS;
    float mnX = pstat[0][0], mxX = pstat[0][1];
    float mnY = pstat[0][2], mxY = pstat[0][3];
    float sumx = pstat[0][4], sumy = pstat[0][5];
    float sqx  = pstat[0][6], sqy  = pstat[0][7];
    float meanx = sumx * invN, meany = sumy * invN;
    float stdx = sqrtf(fmaxf(sqx * invN - meanx * meanx, 0.f));
    float stdy = sqrtf(fmaxf(sqy * invN - meany * meany, 0.f));
    float dmean = d1 * invN;
    float dstd  = sqrtf(fmaxf(d2 * invN - dmean * dmean, 0.f));
    float* g = geo + (size_t)b * GEOPAD;
    g[0] = (mnX + mxX) * 0.5f;  g[1] = (mnY + mxY) * 0.5f;   // bbox center
    g[2] = mxX - mnX;           g[3] = mxY - mnY;            // bbox size
    g[4] = meanx;               g[5] = meany;                // centroid
    g[6] = stdx;                g[7] = stdy;                 // std
    g[8] = dmean;               g[9] = dstd;                 // dist stats
    #pragma unroll
    for (int j = 10; j < GEOPAD; ++j) g[j] = 0.f;            // K-pad
  }
}

// ---------------------------------------------------------------------------
// Kernel 2: zero-pad W1 (10x64) to 16x64 so WMMA K=16 accumulates exactly.
// ---------------------------------------------------------------------------
__global__ void padw1_kernel(const float* __restrict__ W1, float* __restrict__ W1p) {
  int idx = blockIdx.x * blockDim.x + threadIdx.x;   // 0..1023
  int r = idx / HID, c = idx % HID;
  W1p[idx] = (r < 10) ? W1[r * HID + c] : 0.f;
}

// ---------------------------------------------------------------------------
// Kernel 3: fused two-layer MLP via V_WMMA_F32_16X16X4_F32.
// 8 waves/block, each wave owns one 16-row tile (256 tiles = 32 blocks).
// Layer-1 D tile -> LDS (padded stride) -> layer-2 A fragments.
// Fragment layouts per ISA 7.12.2:
//   A 16x4 f32: lanes 0-15 hold M=lane, {K=k0,k0+1}; lanes 16-31 {K=k0+2,k0+3}
//   B 4x16 f32: mirrored (rows striped across lanes, same half-split)
//   C/D 16x16:  VGPR r holds M=r (lanes 0-15) and M=r+8 (lanes 16-31), N=lane%16
// ---------------------------------------------------------------------------
__global__ void __launch_bounds__(TPB)
mlp_kernel(const float* __restrict__ geo, const float* __restrict__ W1p,
           const float* __restrict__ b1, const float* __restrict__ W2,
           const float* __restrict__ b2, float* __restrict__ out) {
  __shared__ float Hs[8][16 * HSTRIDE];   // per-wave 16x64 hidden tile, ~33 KB

  const int lane = threadIdx.x & 31;
  const int wv   = threadIdx.x >> 5;
  const int tile = blockIdx.x * 8 + wv;   // 0..255
  const int row0 = tile * 16;

  const int m    = lane & 15;             // row (A) / col (B,C,D) index
  const int hi   = lane >> 4;             // lane half
  const int koff = hi * 2;

  // ---- Layer 1: (16x16 geo) x (16x64 W1p), K=16 in 4 WMMA k-steps ----
  v8f acc[4];
  #pragma unroll
  for (int nt = 0; nt < 4; ++nt) acc[nt] = (v8f){0.f,0.f,0.f,0.f,0.f,0.f,0.f,0.f};

  #pragma unroll
  for (int k0 = 0; k0 < GEOPAD; k0 += 4) {
    v2f a;
    a.x = geo[(row0 + m) * GEOPAD + k0 + koff];
    a.y = geo[(row0 + m) * GEOPAD + k0 + koff + 1];
    #pragma unroll
    for (int nt = 0; nt < 4; ++nt) {
      v2f bf;
      bf.x = W1p[(k0 + koff)     * HID + nt * 16 + m];
      bf.y = W1p[(k0 + koff + 1) * HID + nt * 16 + m];
      acc[nt] = __builtin_amdgcn_wmma_f32_16x16x4_f32(
          false, a, false, bf, (short)0, acc[nt], false, false);
    }
  }

  // bias + ReLU, scatter D layout into LDS row-major hidden tile
  #pragma unroll
  for (int nt = 0; nt < 4; ++nt) {
    #pragma unroll
    for (int r = 0; r < 8; ++r) {
      int M  = r + hi * 8;
      int Nc = nt * 16 + m;
      Hs[wv][M * HSTRIDE + Nc] = fmaxf(acc[nt][r] + b1[Nc], 0.f);
    }
  }
  __syncthreads();

  // ---- Layer 2: (16x64 H) x (64x32 W2), K=64 in 16 WMMA k-steps ----
  v8f acc2[2];
  acc2[0] = (v8f){0.f,0.f,0.f,0.f,0.f,0.f,0.f,0.f};
  acc2[1] = acc2[0];

  #pragma unroll
  for (int k0 = 0; k0 < HID; k0 += 4) {
    v2f a;
    a.x = Hs[wv][m * HSTRIDE + k0 + koff];
    a.y = Hs[wv][m * HSTRIDE + k0 + koff + 1];
    #pragma unroll
    for (int nt = 0; nt < 2; ++nt) {
      v2f bf;
      bf.x = W2[(k0 + koff)     * OUTD + nt * 16 + m];
      bf.y = W2[(k0 + koff + 1) * OUTD + nt * 16 + m];
      acc2[nt] = __builtin_amdgcn_wmma_f32_16x16x4_f32(
          false, a, false, bf, (short)0, acc2[nt], false, false);
    }
  }

  // bias + ReLU, store 16x32 output tile
  #pragma unroll
  for (int nt = 0; nt < 2; ++nt) {
    #pragma unroll
    for (int r = 0; r < 8; ++r) {
      int M  = r + hi * 8;
      int Nc = nt * 16 + m;
      out[(size_t)(row0 + M) * OUTD + Nc] = fmaxf(acc2[nt][r] + b2[Nc], 0.f);
    }
  }
}

extern "C" void kernel_launch(void* const* d_in, const int* in_sizes, int n_in,
                              void* d_out, int out_size, void* d_ws, size_t ws_size,
                              hipStream_t stream) {
  const float* points = (const float*)d_in[0];   // (4096, 4096, 2)
  const float* W1     = (const float*)d_in[1];   // (10, 64)
  const float* b1     = (const float*)d_in[2];   // (64,)
  const float* W2     = (const float*)d_in[3];   // (64, 32)
  const float* b2     = (const float*)d_in[4];   // (32,)
  float*       out    = (float*)d_out;           // (4096, 32)

  // workspace: geo padded (4096x16 f32 = 256 KB) + W1 padded (16x64 f32 = 4 KB)
  float* geo = (float*)d_ws;
  float* W1p = geo + (size_t)BROWS * GEOPAD;

  geo_kernel<<<BROWS, TPB, 0, stream>>>(points, geo);
  padw1_kernel<<<1, GEOPAD * HID, 0, stream>>>(W1, W1p);
  mlp_kernel<<<(BROWS / 16) / 8, TPB, 0, stream>>>(geo, W1p, b1, W2, b2, out);
}